// TDSAttention_20323785244767
// MI455X (gfx1250) — compile-verified
//
#include <hip/hip_runtime.h>

typedef __attribute__((ext_vector_type(2))) float v2f;
typedef __attribute__((ext_vector_type(8))) float v8f;

constexpr int B_  = 4;
constexpr int C_  = 512;
constexpr int H_  = 96;
constexpr int W_  = 96;
constexpr int HW_ = H_ * W_;   // 9216
constexpr int CR_ = 8;
constexpr float NEG_ = -1e9f;

__device__ __forceinline__ v8f wmma4(v2f a, v2f b, v8f c) {
  // V_WMMA_F32_16X16X4_F32 : D = A(16x4) * B(4x16) + C(16x16), fp32 throughout
  return __builtin_amdgcn_wmma_f32_16x16x4_f32(false, a, false, b, (short)0, c, false, false);
}

__device__ __forceinline__ v8f zero8() {
  v8f v;
#pragma unroll
  for (int i = 0; i < 8; ++i) v[i] = 0.0f;
  return v;
}

// ---------------------------------------------------------------------------
// 1) Cr=8 1x1 convs (f_hei, f_wid). Tiny (0.3 GFLOP each) -> VALU, weights in LDS.
// ---------------------------------------------------------------------------
__global__ void k_smallconv(const float* __restrict__ x,
                            const float* __restrict__ w_h, const float* __restrict__ b_h,
                            const float* __restrict__ w_w, const float* __restrict__ b_w,
                            float* __restrict__ f_hei, float* __restrict__ f_wid) {
  __shared__ float lw[2 * CR_ * C_];               // 32 KB
  const int tid = threadIdx.x;
  for (int i = tid; i < CR_ * C_; i += blockDim.x) {
    lw[i] = w_h[i];
    lw[CR_ * C_ + i] = w_w[i];
  }
  __syncthreads();
  const int p = blockIdx.x * blockDim.x + tid;     // [0, B*HW)
  if (p >= B_ * HW_) return;
  const int b = p / HW_, sp = p % HW_;
  const float* xb = x + (size_t)b * C_ * HW_ + sp;
  float ah[CR_], aw[CR_];
#pragma unroll
  for (int r = 0; r < CR_; ++r) { ah[r] = 0.0f; aw[r] = 0.0f; }
  for (int c = 0; c < C_; ++c) {
    const float v = xb[(size_t)c * HW_];
#pragma unroll
    for (int r = 0; r < CR_; ++r) {
      ah[r] = fmaf(v, lw[r * C_ + c], ah[r]);
      aw[r] = fmaf(v, lw[CR_ * C_ + r * C_ + c], aw[r]);
    }
  }
#pragma unroll
  for (int r = 0; r < CR_; ++r) {
    f_hei[(size_t)(b * CR_ + r) * HW_ + sp] = ah[r] + b_h[r];
    f_wid[(size_t)(b * CR_ + r) * HW_ + sp] = aw[r] + b_w[r];
  }
}

// ---------------------------------------------------------------------------
// 2) C->C 1x1 conv: Y[b,o,p] = sum_c W[o,c]*X[b,c,p] + bias[o].
//    WMMA f32 16x16x4; wave computes a 16(o) x 64(p) tile; bias in C operand.
//    grid = (HW/64, C/64, B), block = 128 (4 waves).
// ---------------------------------------------------------------------------
__global__ void k_conv_cc(const float* __restrict__ x, const float* __restrict__ w,
                          const float* __restrict__ bias, float* __restrict__ y) {
  const int lane = threadIdx.x & 31;
  const int wv   = threadIdx.x >> 5;
  const int b    = blockIdx.z;
  const int o0   = (blockIdx.y * 4 + wv) * 16;
  const int p0   = blockIdx.x * 64;
  const int m    = lane & 15;
  const int half = lane >> 4;
  const int koff = half * 2;
  const float* xb = x + (size_t)b * C_ * HW_;
  v8f acc0, acc1, acc2, acc3;
#pragma unroll
  for (int r = 0; r < 8; ++r) {               // C/D layout: VGPR r = row (r + 8*half)
    const float bv = bias[o0 + r + 8 * half];
    acc0[r] = bv; acc1[r] = bv; acc2[r] = bv; acc3[r] = bv;
  }
  for (int c = 0; c < C_; c += 4) {
    const v2f a = *(const v2f*)(w + (o0 + m) * C_ + c + koff);   // A: K pair contiguous
    const float* x0 = xb + (size_t)(c + koff) * HW_ + p0 + m;    // B: K=koff row
    const float* x1 = x0 + HW_;                                  //    K=koff+1 row
    const v2f b0{x0[0],  x1[0]};
    const v2f b1{x0[16], x1[16]};
    const v2f b2{x0[32], x1[32]};
    const v2f b3{x0[48], x1[48]};
    acc0 = wmma4(a, b0, acc0);
    acc1 = wmma4(a, b1, acc1);
    acc2 = wmma4(a, b2, acc2);
    acc3 = wmma4(a, b3, acc3);
  }
  float* yb = y + (size_t)b * C_ * HW_;
#pragma unroll
  for (int r = 0; r < 8; ++r) {
    const int row = o0 + r + 8 * half;
    float* yr = yb + (size_t)row * HW_ + p0 + m;
    yr[0] = acc0[r]; yr[16] = acc1[r]; yr[32] = acc2[r]; yr[48] = acc3[r];
  }
}

// ---------------------------------------------------------------------------
// 3) Gram: S[b,i,j] = sum_p F[b,i,p]*F[b,j,p].  Both fragments are contiguous
//    8B loads along p. Wave: 16(i) x 64(j), K=HW. grid=(C/64, C/64, B), block=128.
// ---------------------------------------------------------------------------
__global__ void k_gram(const float* __restrict__ f, float* __restrict__ S) {
  const int lane = threadIdx.x & 31;
  const int wv   = threadIdx.x >> 5;
  const int b    = blockIdx.z;
  const int i0   = (blockIdx.y * 4 + wv) * 16;
  const int j0   = blockIdx.x * 64;
  const int m    = lane & 15;
  const int half = lane >> 4;
  const int koff = half * 2;
  const float* fb = f + (size_t)b * C_ * HW_;
  v8f acc0 = zero8(), acc1 = zero8(), acc2 = zero8(), acc3 = zero8();
  const float* ar  = fb + (size_t)(i0 + m) * HW_ + koff;
  const float* br0 = fb + (size_t)(j0 + m) * HW_ + koff;
  const float* br1 = br0 + (size_t)16 * HW_;
  const float* br2 = br0 + (size_t)32 * HW_;
  const float* br3 = br0 + (size_t)48 * HW_;
  for (int p = 0; p < HW_; p += 4) {
    const v2f a = *(const v2f*)(ar + p);
    acc0 = wmma4(a, *(const v2f*)(br0 + p), acc0);
    acc1 = wmma4(a, *(const v2f*)(br1 + p), acc1);
    acc2 = wmma4(a, *(const v2f*)(br2 + p), acc2);
    acc3 = wmma4(a, *(const v2f*)(br3 + p), acc3);
  }
  float* Sb = S + (size_t)b * C_ * C_;
#pragma unroll
  for (int r = 0; r < 8; ++r) {
    const int row = i0 + r + 8 * half;
    float* sr = Sb + (size_t)row * C_ + j0 + m;
    sr[0] = acc0[r]; sr[16] = acc1[r]; sr[32] = acc2[r]; sr[48] = acc3[r];
  }
}

// ---------------------------------------------------------------------------
// 4) Row softmax over 512 (in place). grid = B*C rows, block = 256.
// ---------------------------------------------------------------------------
__global__ void k_softmax512(float* __restrict__ S) {
  float* r = S + (size_t)blockIdx.x * C_;
  const int t = threadIdx.x;
  __shared__ float red[256];
  const float v0 = r[t], v1 = r[t + 256];
  red[t] = fmaxf(v0, v1);
  __syncthreads();
  for (int s = 128; s > 0; s >>= 1) { if (t < s) red[t] = fmaxf(red[t], red[t + s]); __syncthreads(); }
  const float mx = red[0];
  __syncthreads();
  const float e0 = expf(v0 - mx), e1 = expf(v1 - mx);
  red[t] = e0 + e1;
  __syncthreads();
  for (int s = 128; s > 0; s >>= 1) { if (t < s) red[t] += red[t + s]; __syncthreads(); }
  const float inv = 1.0f / red[0];
  r[t] = e0 * inv;
  r[t + 256] = e1 * inv;
}

// ---------------------------------------------------------------------------
// 5) out_C GEMM: D[b,p,j] = sum_i F[b,i,p]*aff[b,i,j], written gamma-scaled
//    into d_out ([B,HW,C]-flat == the reference's .view as [B,C,H,W]-flat).
//    Wave: 16(p) x 64(j), K=C. grid=(HW/64, C/64, B), block=128.
// ---------------------------------------------------------------------------
__global__ void k_outc(const float* __restrict__ f, const float* __restrict__ aff,
                       const float* __restrict__ gp, float* __restrict__ out) {
  const int lane = threadIdx.x & 31;
  const int wv   = threadIdx.x >> 5;
  const int b    = blockIdx.z;
  const int p0   = (blockIdx.x * 4 + wv) * 16;
  const int j0   = blockIdx.y * 64;
  const int m    = lane & 15;
  const int half = lane >> 4;
  const int koff = half * 2;
  const float* fb = f + (size_t)b * C_ * HW_;
  const float* ab = aff + (size_t)b * C_ * C_;
  v8f acc0 = zero8(), acc1 = zero8(), acc2 = zero8(), acc3 = zero8();
  for (int i = 0; i < C_; i += 4) {
    const v2f a{fb[(size_t)(i + koff) * HW_ + p0 + m],
                fb[(size_t)(i + koff + 1) * HW_ + p0 + m]};
    const float* a0 = ab + (size_t)(i + koff) * C_ + j0 + m;
    const float* a1 = a0 + C_;
    const v2f b0{a0[0],  a1[0]};
    const v2f b1{a0[16], a1[16]};
    const v2f b2{a0[32], a1[32]};
    const v2f b3{a0[48], a1[48]};
    acc0 = wmma4(a, b0, acc0);
    acc1 = wmma4(a, b1, acc1);
    acc2 = wmma4(a, b2, acc2);
    acc3 = wmma4(a, b3, acc3);
  }
  const float g = gp[0];
  float* ob = out + (size_t)b * HW_ * C_;
#pragma unroll
  for (int r = 0; r < 8; ++r) {
    const int prow = p0 + r + 8 * half;
    float* orow = ob + (size_t)prow * C_ + j0 + m;
    orow[0]  = g * acc0[r];
    orow[16] = g * acc1[r];
    orow[32] = g * acc2[r];
    orow[48] = g * acc3[r];
  }
}

// ---------------------------------------------------------------------------
// 6) Axial energies (K=Cr=8, includes the torch .view gather on f_h) + masked
//    concat-softmax over H+W=192.  One block per (b,y,x), 192 threads.
// ---------------------------------------------------------------------------
__global__ void k_att(const float* __restrict__ f_hei, const float* __restrict__ f_wid,
                      float* __restrict__ att) {
  const int idx = blockIdx.x;               // ((b*H + y)*W + x)
  const int xw  = idx % W_;
  const int y   = (idx / W_) % H_;
  const int b   = idx / (W_ * H_);
  const int j   = threadIdx.x;              // 0..191
  const float* fh = f_hei + (size_t)b * CR_ * HW_;
  const float* fw = f_wid + (size_t)b * CR_ * HW_;
  float e;
  if (j < H_) {
    // E_h: f_h[(b,x), y, a] = f_hei[b, y&7, 12a + (y>>3), x] (view reinterpretation)
    const int rr = y & 7, hb = y >> 3;
    float s = 0.0f;
#pragma unroll
    for (int a = 0; a < CR_; ++a)
      s = fmaf(fh[rr * HW_ + (12 * a + hb) * W_ + xw], fw[a * HW_ + j * W_ + xw], s);
    if (j == y) s += NEG_;
    e = s;
  } else {
    const int jj = j - H_;
    float s = 0.0f;
#pragma unroll
    for (int a = 0; a < CR_; ++a)
      s = fmaf(fh[a * HW_ + y * W_ + xw], fw[a * HW_ + y * W_ + jj], s);
    e = s;
  }
  __shared__ float red[192];
  __shared__ float r2[64];
  red[j] = e;
  __syncthreads();
  if (j < 64) r2[j] = fmaxf(red[j], fmaxf(red[j + 64], red[j + 128]));
  __syncthreads();
  for (int s = 32; s > 0; s >>= 1) { if (j < s) r2[j] = fmaxf(r2[j], r2[j + s]); __syncthreads(); }
  const float mx = r2[0];
  __syncthreads();
  const float ee = expf(e - mx);
  red[j] = ee;
  __syncthreads();
  if (j < 64) r2[j] = red[j] + red[j + 64] + red[j + 128];
  __syncthreads();
  for (int s = 32; s > 0; s >>= 1) { if (j < s) r2[j] += r2[j + s]; __syncthreads(); }
  att[(size_t)idx * 192 + j] = ee / r2[0];
}

// ---------------------------------------------------------------------------
// 7) Fused final: out = x + alpha*out_H + beta*out_W + (out == gamma*out_C).
//    Block per (b,y); thread = x column; att rows staged in LDS (stride 97).
// ---------------------------------------------------------------------------
__global__ void k_final(const float* __restrict__ x, const float* __restrict__ f_cha,
                        const float* __restrict__ att, const float* __restrict__ ap,
                        const float* __restrict__ bp, float* __restrict__ out) {
  const int by = blockIdx.x;                // b*H + y
  const int y  = by % H_;
  const int b  = by / H_;
  const int xw = threadIdx.x;               // 0..95
  extern __shared__ float sm[];             // 2 * 96 * 97 floats
  float* atth = sm;                         // atth[x][y'] (stride 97)
  float* attw = sm + 96 * 97;               // attw[x][x'] (stride 97)
  const float* arow = att + (size_t)(b * H_ + y) * W_ * 192;
  for (int q = xw; q < W_ * 192; q += W_) {
    const int xx = q / 192, j = q % 192;
    const float v = arow[q];
    if (j < H_) atth[xx * 97 + j] = v;
    else        attw[xx * 97 + (j - H_)] = v;
  }
  __syncthreads();
  const float alpha = ap[0], beta = bp[0];
  const float* fb = f_cha + (size_t)b * C_ * HW_;
  const float* xb = x + (size_t)b * C_ * HW_;
  float* ob = out + (size_t)b * C_ * HW_;
  const float* ah = atth + xw * 97;
  const float* aw = attw + xw * 97;
  for (int c = 0; c < C_; ++c) {
    const float* pc = fb + (size_t)c * HW_;
    float sH = 0.0f, sW = 0.0f;
#pragma unroll 4
    for (int t = 0; t < 96; ++t) {
      sH = fmaf(pc[t * W_ + xw], ah[t], sH);   // out_H: column gather, coalesced across x
      sW = fmaf(pc[y * W_ + t], aw[t], sW);    // out_W: row broadcast
    }
    const size_t i = (size_t)c * HW_ + y * W_ + xw;
    ob[i] = xb[i] + alpha * sH + beta * sW + ob[i];
  }
}

// ---------------------------------------------------------------------------
extern "C" void kernel_launch(void* const* d_in, const int* in_sizes, int n_in,
                              void* d_out, int out_size, void* d_ws, size_t ws_size,
                              hipStream_t stream) {
  (void)in_sizes; (void)n_in; (void)out_size; (void)ws_size;
  const float* x     = (const float*)d_in[0];
  const float* w_h   = (const float*)d_in[1];
  const float* b_h   = (const float*)d_in[2];
  const float* w_w   = (const float*)d_in[3];
  const float* b_w   = (const float*)d_in[4];
  const float* w_c   = (const float*)d_in[5];
  const float* b_c   = (const float*)d_in[6];
  const float* w_o   = (const float*)d_in[7];
  const float* b_o   = (const float*)d_in[8];
  const float* alpha = (const float*)d_in[9];
  const float* beta  = (const float*)d_in[10];
  const float* gamma = (const float*)d_in[11];
  float* out = (float*)d_out;

  float* ws    = (float*)d_ws;
  float* f_ch  = ws;                                   // [B,C,HW]   18.87M f
  float* f_cha = f_ch  + (size_t)B_ * C_ * HW_;        // [B,C,HW]   18.87M f
  float* S     = f_cha + (size_t)B_ * C_ * HW_;        // [B,C,C]     1.05M f
  float* f_hei = S     + (size_t)B_ * C_ * C_;         // [B,Cr,H,W]  0.29M f
  float* f_wid = f_hei + (size_t)B_ * CR_ * HW_;       // [B,Cr,H,W]  0.29M f
  float* att   = f_wid + (size_t)B_ * CR_ * HW_;       // [B,H,W,192] 7.08M f

  k_smallconv<<<dim3((B_ * HW_ + 255) / 256), dim3(256), 0, stream>>>(
      x, w_h, b_h, w_w, b_w, f_hei, f_wid);
  k_conv_cc<<<dim3(HW_ / 64, C_ / 64, B_), dim3(128), 0, stream>>>(x, w_c, b_c, f_ch);
  k_conv_cc<<<dim3(HW_ / 64, C_ / 64, B_), dim3(128), 0, stream>>>(x, w_o, b_o, f_cha);
  k_gram<<<dim3(C_ / 64, C_ / 64, B_), dim3(128), 0, stream>>>(f_ch, S);
  k_softmax512<<<dim3(B_ * C_), dim3(256), 0, stream>>>(S);
  k_outc<<<dim3(HW_ / 64, C_ / 64, B_), dim3(128), 0, stream>>>(f_ch, S, gamma, out);
  k_att<<<dim3(B_ * H_ * W_), dim3(192), 0, stream>>>(f_hei, f_wid, att);
  k_final<<<dim3(B_ * H_), dim3(96), 2 * 96 * 97 * sizeof(float), stream>>>(
      x, f_cha, att, alpha, beta, out);
}